// Net_21706764714346
// MI455X (gfx1250) — compile-verified
//
#include <hip/hip_runtime.h>
#include <hip/hip_bf16.h>

typedef __attribute__((ext_vector_type(2))) float v2f;
typedef __attribute__((ext_vector_type(8))) float v8f;

#define N_NODES   100000
#define N_EDGES   3200000
#define D_FEAT    128
#define HIDDEN    16
#define N_CLASSES 7
#define NCHUNK    (D_FEAT / 4)   // 32 K-steps of 4

// ---------------------------------------------------------------------------
// Layer-1 dense projection: h1 = x @ W1  ([100000,128] x [128,16])
// One wave (wave32) computes 16x16 output tiles via chained
// V_WMMA_F32_16X16X4_F32 (fp32-exact matrix pipe). The B operand (W1) is
// tile-invariant, so all 32 B fragments are hoisted into registers once;
// the inner loop is then one global_load_b64 (A) + one v_wmma per K-step.
//
// fp32 A (16x4) layout:  lanes 0-15 / 16-31 both hold M = lane&15;
//   VGPR0 = K + 2*(lane>=16), VGPR1 = that + 1  -> contiguous 8B per lane.
// fp32 B (4x16) layout:  VGPR0 = row K+2*(lane>=16), col = lane&15; VGPR1 = +1.
// fp32 C/D (16x16): VGPR r = row r (lanes 0-15) / row r+8 (lanes 16-31).
// ---------------------------------------------------------------------------
__global__ void gemm1_wmma(const float* __restrict__ x,
                           const float* __restrict__ W1,
                           float* __restrict__ h1) {
    const int lane = threadIdx.x & 31;          // wave32 (gfx1250)
    const int wave = threadIdx.x >> 5;
    const int m    = lane & 15;
    const int hi   = lane >> 4;                 // 0: lanes 0-15, 1: lanes 16-31

    // Hoisted B fragments: W1 is 8KB, loaded once per wave.
    v2f bfrag[NCHUNK];
#pragma unroll
    for (int kk = 0; kk < NCHUNK; ++kk) {
        const int k0 = kk * 4 + 2 * hi;
        bfrag[kk].x = W1[k0 * HIDDEN + m];
        bfrag[kk].y = W1[(k0 + 1) * HIDDEN + m];
    }

    const int ntiles = N_NODES / 16;            // 6250, exact
    const int nwaves = gridDim.x * (blockDim.x >> 5);

    for (int tile = blockIdx.x * (blockDim.x >> 5) + wave; tile < ntiles;
         tile += nwaves) {                      // wave-uniform: EXEC stays all-1s
        const float* xrow = x + (size_t)(tile * 16 + m) * D_FEAT;

        // Prefetch this wave's next tile row (speculative; OOB is dropped).
        __builtin_prefetch(xrow + (size_t)nwaves * 16 * D_FEAT, 0, 0);

        v8f c = {0.f, 0.f, 0.f, 0.f, 0.f, 0.f, 0.f, 0.f};
#pragma unroll
        for (int kk = 0; kk < NCHUNK; ++kk) {
            const int k0 = kk * 4 + 2 * hi;
            v2f a = *(const v2f*)(xrow + k0);   // 8B-aligned global_load_b64
            c = __builtin_amdgcn_wmma_f32_16x16x4_f32(
                    /*neg_a=*/false, a, /*neg_b=*/false, bfrag[kk],
                    /*c_mod=*/(short)0, c, /*reuse_a=*/false, /*reuse_b=*/false);
        }

#pragma unroll
        for (int r = 0; r < 8; ++r) {
            const int orow = tile * 16 + r + hi * 8;
            h1[(size_t)orow * HIDDEN + m] = c[r];
        }
    }
}

// ---------------------------------------------------------------------------
// Degree / normalization
// ---------------------------------------------------------------------------
__global__ void deg_init(float* __restrict__ deg) {
    int i = blockIdx.x * blockDim.x + threadIdx.x;
    if (i < N_NODES) deg[i] = 1.0f;             // self-loop weight
}

__global__ void deg_scatter(const long long* __restrict__ ei,
                            const float* __restrict__ ew,
                            float* __restrict__ deg) {
    int e = blockIdx.x * blockDim.x + threadIdx.x;
    if (e < N_EDGES) {
        long long d = ei[(size_t)N_EDGES + e];  // dst row of edge_index
        atomicAdd(&deg[d], ew[e]);
    }
}

__global__ void dinv_kernel(const float* __restrict__ deg, float* __restrict__ dinv) {
    int i = blockIdx.x * blockDim.x + threadIdx.x;
    if (i < N_NODES) {
        float v = deg[i];
        dinv[i] = (v > 0.f) ? rsqrtf(v) : 0.f;
    }
}

__global__ void zero_f32(float* __restrict__ p, int n) {
    int i = blockIdx.x * blockDim.x + threadIdx.x;
    if (i < n) p[i] = 0.f;
}

// ---------------------------------------------------------------------------
// Layer-1 message passing: 4 lanes per edge, float4 gathers, 4 atomics each.
// h1/out1 (6.4 MB each) are L2-resident; atomics resolve in L2 atomic units.
// ---------------------------------------------------------------------------
__global__ void scatter1(const long long* __restrict__ ei,
                         const float* __restrict__ ew,
                         const float* __restrict__ dinv,
                         const float* __restrict__ h1,
                         float* __restrict__ out1) {
    unsigned idx = blockIdx.x * blockDim.x + threadIdx.x;
    if (idx >= (unsigned)N_EDGES * 4u) return;
    const unsigned e  = idx >> 2;
    const unsigned f0 = (idx & 3u) * 4u;        // features f0 .. f0+3
    const long long s = ei[e];
    const long long d = ei[(size_t)N_EDGES + e];
    const float nrm = dinv[s] * ew[e] * dinv[d];
    const float4 hv = *(const float4*)(h1 + (size_t)s * HIDDEN + f0);
    float* o = out1 + (size_t)d * HIDDEN + f0;
    atomicAdd(o + 0, hv.x * nrm);
    atomicAdd(o + 1, hv.y * nrm);
    atomicAdd(o + 2, hv.z * nrm);
    atomicAdd(o + 3, hv.w * nrm);
}

// out1 = relu(out1 + h1 * dinv^2 (self loop) + b1), float4 per thread
__global__ void finalize1(const float* __restrict__ h1,
                          const float* __restrict__ dinv,
                          const float* __restrict__ b1,
                          float* __restrict__ out1) {
    unsigned idx = blockIdx.x * blockDim.x + threadIdx.x;     // (node, 4-feat group)
    if (idx >= (unsigned)N_NODES * (HIDDEN / 4)) return;
    const unsigned i  = idx >> 2;
    const unsigned f0 = (idx & 3u) * 4u;
    const float di  = dinv[i];
    const float d2  = di * di;
    const float4 hv = *(const float4*)(h1 + (size_t)i * HIDDEN + f0);
    const float4 bv = *(const float4*)(b1 + f0);
    float4 ov = *(float4*)(out1 + (size_t)i * HIDDEN + f0);
    ov.x = fmaxf(ov.x + hv.x * d2 + bv.x, 0.f);
    ov.y = fmaxf(ov.y + hv.y * d2 + bv.y, 0.f);
    ov.z = fmaxf(ov.z + hv.z * d2 + bv.z, 0.f);
    ov.w = fmaxf(ov.w + hv.w * d2 + bv.w, 0.f);
    *(float4*)(out1 + (size_t)i * HIDDEN + f0) = ov;
}

// ---------------------------------------------------------------------------
// Layer-2 projection: h2 = out1 @ W2  (K=16, N=7 -- scalar VALU is right-sized)
// ---------------------------------------------------------------------------
__global__ void gemm2(const float* __restrict__ out1,
                      const float* __restrict__ W2,
                      float* __restrict__ h2) {
    int i = blockIdx.x * blockDim.x + threadIdx.x;
    if (i >= N_NODES) return;
    float a[HIDDEN];
#pragma unroll
    for (int k = 0; k < HIDDEN; ++k) a[k] = out1[(size_t)i * HIDDEN + k];
#pragma unroll
    for (int c = 0; c < N_CLASSES; ++c) {
        float acc = 0.f;
#pragma unroll
        for (int k = 0; k < HIDDEN; ++k) acc = fmaf(a[k], W2[k * N_CLASSES + c], acc);
        h2[(size_t)i * N_CLASSES + c] = acc;
    }
}

// Layer-2 message passing: 1 thread per edge, 7 features
__global__ void scatter2(const long long* __restrict__ ei,
                         const float* __restrict__ ew,
                         const float* __restrict__ dinv,
                         const float* __restrict__ h2,
                         float* __restrict__ out2) {
    int e = blockIdx.x * blockDim.x + threadIdx.x;
    if (e >= N_EDGES) return;
    const long long s = ei[e];
    const long long d = ei[(size_t)N_EDGES + e];
    const float nrm = dinv[s] * ew[e] * dinv[d];
    const float* hs = h2 + (size_t)s * N_CLASSES;
    float* o = out2 + (size_t)d * N_CLASSES;
#pragma unroll
    for (int c = 0; c < N_CLASSES; ++c) atomicAdd(o + c, hs[c] * nrm);
}

// out = log_softmax(out2 + h2 * dinv^2 + b2) per node, in place on d_out
__global__ void finalize2(const float* __restrict__ h2,
                          const float* __restrict__ dinv,
                          const float* __restrict__ b2,
                          float* __restrict__ out) {
    int i = blockIdx.x * blockDim.x + threadIdx.x;
    if (i >= N_NODES) return;
    const float di = dinv[i];
    float v[N_CLASSES];
    float m = -3.402823466e+38f;
#pragma unroll
    for (int c = 0; c < N_CLASSES; ++c) {
        v[c] = out[(size_t)i * N_CLASSES + c] + h2[(size_t)i * N_CLASSES + c] * di * di + b2[c];
        m = fmaxf(m, v[c]);
    }
    float s = 0.f;
#pragma unroll
    for (int c = 0; c < N_CLASSES; ++c) s += expf(v[c] - m);
    const float lse = m + logf(s);
#pragma unroll
    for (int c = 0; c < N_CLASSES; ++c) out[(size_t)i * N_CLASSES + c] = v[c] - lse;
}

// ---------------------------------------------------------------------------
extern "C" void kernel_launch(void* const* d_in, const int* in_sizes, int n_in,
                              void* d_out, int out_size, void* d_ws, size_t ws_size,
                              hipStream_t stream) {
    const float*     x   = (const float*)d_in[0];
    const long long* ei  = (const long long*)d_in[1];  // int64 [2, E]
    const float*     ew  = (const float*)d_in[2];
    const float*     W1  = (const float*)d_in[3];
    const float*     b1  = (const float*)d_in[4];
    const float*     W2  = (const float*)d_in[5];
    const float*     b2  = (const float*)d_in[6];
    float* out = (float*)d_out;

    // Workspace layout (fp32)
    float* h1   = (float*)d_ws;                     // N_NODES*16 = 1.6M
    float* out1 = h1   + (size_t)N_NODES * HIDDEN;  // 1.6M
    float* deg  = out1 + (size_t)N_NODES * HIDDEN;  // 100K
    float* dinv = deg  + N_NODES;                   // 100K
    float* h2   = dinv + N_NODES;                   // 700K
    // total ~16.4 MB

    const int B = 256;

    // Dense projection (WMMA fp32)
    {
        const int ntiles = N_NODES / 16;            // 6250
        const int waves_per_block = B / 32;         // 8
        const int blocks = (ntiles + waves_per_block - 1) / waves_per_block;
        hipLaunchKernelGGL(gemm1_wmma, dim3(blocks), dim3(B), 0, stream, x, W1, h1);
    }

    // Degree + normalization
    hipLaunchKernelGGL(deg_init,    dim3((N_NODES + B - 1) / B), dim3(B), 0, stream, deg);
    hipLaunchKernelGGL(deg_scatter, dim3((N_EDGES + B - 1) / B), dim3(B), 0, stream, ei, ew, deg);
    hipLaunchKernelGGL(dinv_kernel, dim3((N_NODES + B - 1) / B), dim3(B), 0, stream, deg, dinv);

    // Layer 1 propagation
    {
        const unsigned n = (unsigned)N_NODES * HIDDEN;
        hipLaunchKernelGGL(zero_f32, dim3((n + B - 1) / B), dim3(B), 0, stream, out1, (int)n);
        const unsigned t = (unsigned)N_EDGES * 4u;
        hipLaunchKernelGGL(scatter1, dim3((t + B - 1) / B), dim3(B), 0, stream, ei, ew, dinv, h1, out1);
        const unsigned g = (unsigned)N_NODES * (HIDDEN / 4);
        hipLaunchKernelGGL(finalize1, dim3((g + B - 1) / B), dim3(B), 0, stream, h1, dinv, b1, out1);
    }

    // Layer 2 projection + propagation
    hipLaunchKernelGGL(gemm2, dim3((N_NODES + B - 1) / B), dim3(B), 0, stream, out1, W2, h2);
    {
        const int n = N_NODES * N_CLASSES;
        hipLaunchKernelGGL(zero_f32, dim3((n + B - 1) / B), dim3(B), 0, stream, out, n);
        hipLaunchKernelGGL(scatter2, dim3((N_EDGES + B - 1) / B), dim3(B), 0, stream, ei, ew, dinv, h2, out);
        hipLaunchKernelGGL(finalize2, dim3((N_NODES + B - 1) / B), dim3(B), 0, stream, h2, dinv, b2, out);
    }
}